// QConv2d_14302241096055
// MI455X (gfx1250) — compile-verified
//
#include <hip/hip_runtime.h>
#include <hip/hip_bf16.h>

typedef float v2f __attribute__((ext_vector_type(2)));
typedef float v4f __attribute__((ext_vector_type(4)));
typedef float v8f __attribute__((ext_vector_type(8)));

#define WMMA_F32(A, B, C) \
    __builtin_amdgcn_wmma_f32_16x16x4_f32(false, (A), false, (B), (short)0, (C), false, false)

// WGP-scope prefetch (scope 0 = pull into all cache levels, stop on WGP hit).
// __builtin_prefetch lowers to scope:SCOPE_SE which skips the WGP cache.
__device__ __forceinline__ void prefetch_wgp(const void* p) {
    asm volatile("global_prefetch_b8 %0, off" :: "v"(p));
}

// ---------------------------------------------------------------------------
// Kernel 1: build U = kron(Ux, Uy) (256x256 complex) and UH = conj(U)^T
// from the 2x56 MZI phases. Tiny one-block kernel; threads 0/1 build the two
// 8x8 filter unitaries sequentially, then all 256 threads form the kron.
// Workspace layout (floats): Ur[65536] | Ui[65536] | UHr[65536] | UHi[65536]
// ---------------------------------------------------------------------------
__global__ __launch_bounds__(256)
void qconv_build_u(const float* __restrict__ phi_x, const float* __restrict__ phi_y,
                   float* __restrict__ Ur, float* __restrict__ Ui,
                   float* __restrict__ UHr, float* __restrict__ UHi)
{
    __shared__ float ufr[2][64];
    __shared__ float ufi[2][64];

    const int tid = threadIdx.x;
    if (tid < 2) {
        const float* phi = (tid == 0) ? phi_x : phi_y;
        float* wr = ufr[tid];
        float* wi = ufi[tid];
        for (int i = 0; i < 64; ++i) { wr[i] = 0.f; wi[i] = 0.f; }
        for (int i = 0; i < 8; ++i)  { wr[i * 8 + i] = 1.f; }

        int p = 0;
        for (int layer = 0; layer < 8; ++layer) {
            for (int j = (layer & 1); j < 7; j += 2) {
                float pa = phi[p], pb = phi[p + 1]; p += 2;
                float sa, ca, sb, cb;
                __sincosf(pa, &sa, &ca);
                __sincosf(pb, &sb, &cb);
                // M = BS psb BS psa = 0.5*[[ea(eb-1), i(eb+1)],[i ea (eb+1), (1-eb)]]
                float ear = ca, eai = sa;
                float ebr = cb, ebi = sb;
                float em1r = ebr - 1.f, em1i = ebi;
                float ep1r = ebr + 1.f, ep1i = ebi;
                float M00r = 0.5f * (ear * em1r - eai * em1i);
                float M00i = 0.5f * (ear * em1i + eai * em1r);
                float M01r = -0.5f * ep1i;
                float M01i =  0.5f * ep1r;
                float t_r = 0.5f * (ear * ep1r - eai * ep1i);
                float t_i = 0.5f * (ear * ep1i + eai * ep1r);
                float M10r = -t_i;
                float M10i =  t_r;
                float M11r = 0.5f * (1.f - ebr);
                float M11i = -0.5f * ebi;

                for (int c = 0; c < 8; ++c) {
                    float x0r = wr[j * 8 + c],       x0i = wi[j * 8 + c];
                    float x1r = wr[(j + 1) * 8 + c], x1i = wi[(j + 1) * 8 + c];
                    float n0r = M00r * x0r - M00i * x0i + M01r * x1r - M01i * x1i;
                    float n0i = M00r * x0i + M00i * x0r + M01r * x1i + M01i * x1r;
                    float n1r = M10r * x0r - M10i * x0i + M11r * x1r - M11i * x1i;
                    float n1i = M10r * x0i + M10i * x0r + M11r * x1i + M11i * x1r;
                    wr[j * 8 + c] = n0r;       wi[j * 8 + c] = n0i;
                    wr[(j + 1) * 8 + c] = n1r; wi[(j + 1) * 8 + c] = n1i;
                }
            }
        }
    }
    __syncthreads();

    for (int idx = tid; idx < 65536; idx += 256) {
        int row = idx >> 8, col = idx & 255;
        int ix = row >> 4, iy = row & 15;
        int jx = col >> 4, jy = col & 15;
        float xr = 0.f, xi = 0.f;
        if ((ix >> 3) == (jx >> 3)) {
            int o = (ix & 7) * 8 + (jx & 7);
            xr = ufr[0][o]; xi = ufi[0][o];
        }
        float yr = 0.f, yi = 0.f;
        if ((iy >> 3) == (jy >> 3)) {
            int o = (iy & 7) * 8 + (jy & 7);
            yr = ufr[1][o]; yi = ufi[1][o];
        }
        float ur = xr * yr - xi * yi;
        float ui = xr * yi + xi * yr;
        Ur[idx] = ur;
        Ui[idx] = ui;
        UHr[col * 256 + row] = ur;     // UH[k][j] = conj(U[j][k])
        UHi[col * 256 + row] = -ui;
    }
}

// ---------------------------------------------------------------------------
// Kernel 2: fused  out_b = U * rho_b * U^H  (complex, f32 WMMA)
// grid = (4 row-blocks of 64 rows, 256 batches), block = 256 threads (8 waves)
// 64-row blocks cut rho_b L2 re-reads to 4x: ~536 MB total L2 traffic,
// below the 11.5us HBM floor (268 MB @ 23.3 TB/s).
// LDS (dynamic, 266240 B < 320 KB WGP limit):
//   sUr/sUi[64][260] : U row block (stage-1 A operand)
//   sTr/sTi[64][260] : temp = Ublk @ rho_b (stage-1 D, stage-2 A)
// Wave w: row tile rT = w>>1, column tiles (w&1)*8 .. +7 (8 tiles/stage).
// ---------------------------------------------------------------------------
#define RB     64      // rows per workgroup
#define PITCH  260     // 256 + 4 pad floats: distinct LDS banks, 16B aligned

__global__ __launch_bounds__(256)
void qconv_apply(const float* __restrict__ rho_r, const float* __restrict__ rho_i,
                 const float* __restrict__ Ur,  const float* __restrict__ Ui,
                 const float* __restrict__ UHr, const float* __restrict__ UHi,
                 float* __restrict__ out)
{
    const int rblk = blockIdx.x;      // 0..3 : 64-row block of output
    const int b    = blockIdx.y;      // 0..255 : batch
    const int tid  = threadIdx.x;
    const int lane = tid & 31;
    const int wave = tid >> 5;
    const int half = lane >> 4;       // 0: K pair {0,1}, 1: K pair {2,3}
    const int l16  = lane & 15;

    extern __shared__ float smem[];
    float* sUr = smem;                     // RB*PITCH
    float* sUi = sUr + RB * PITCH;
    float* sTr = sUi + RB * PITCH;
    float* sTi = sTr + RB * PITCH;

    // Stage 0: stage the 64-row U block into LDS with b128 loads/stores
    {
        const v4f* gr = (const v4f*)(Ur + rblk * RB * 256);
        const v4f* gi = (const v4f*)(Ui + rblk * RB * 256);
        for (int c = tid; c < RB * 64; c += 256) {      // 64 float4 per row
            int row = c >> 6;
            int off = (c & 63) * 4;
            *(v4f*)&sUr[row * PITCH + off] = gr[c];
            *(v4f*)&sUi[row * PITCH + off] = gi[c];
        }
    }
    __syncthreads();

    const float* rr = rho_r + (size_t)b * 65536;
    const float* ri = rho_i + (size_t)b * 65536;

    const int rT    = wave >> 1;            // row tile within block: 0..3
    const int cbase = (wave & 1) * 8;       // column tile group
    const int arow  = (rT * 16 + l16) * PITCH + half * 2;

    // ---------------- Stage 1: temp = Ublk @ rho_b ----------------
    for (int tt = 0; tt < 8; ++tt) {
        const int cT  = cbase + tt;
        const int col = cT * 16 + l16;
        v8f cr = {};
        v8f ci = {};
        // 32-bit element-offset induction -> saddr + scaled voffset loads
        unsigned o = (unsigned)(half * 512 + col);
        for (int k0 = 0; k0 < 256; k0 += 4, o += 1024) {
            // A operand (16x4 f32) from LDS: lane = row M, half selects K pair
            v2f ar = *(const v2f*)&sUr[arow + k0];
            v2f ai = *(const v2f*)&sUi[arow + k0];
            v2f nai = -ai;
            // B operand (4x16 f32) from global rho (L2-resident)
            v2f br, bi;
            br.x = rr[o];
            br.y = rr[o + 256];
            bi.x = ri[o];
            bi.y = ri[o + 256];
            // pull rho lines 32 K-rows ahead, L2 -> WGP cache
            if (k0 < 224) {
                prefetch_wgp(rr + o + 8192);
                prefetch_wgp(ri + o + 8192);
            }
            cr = WMMA_F32(ar,  br, cr);
            cr = WMMA_F32(nai, bi, cr);
            ci = WMMA_F32(ar,  bi, ci);
            ci = WMMA_F32(ai,  br, ci);
        }
        // scatter C tile (16x16 f32 layout) into temp LDS
        for (int v = 0; v < 8; ++v) {
            const int m = rT * 16 + v + half * 8;
            const int n = cT * 16 + l16;
            sTr[m * PITCH + n] = cr[v];
            sTi[m * PITCH + n] = ci[v];
        }
    }
    __syncthreads();

    // ---------------- Stage 2: out rows = temp @ UH ----------------
    v2f* obase = (v2f*)out + ((size_t)b * 256 + rblk * RB) * 256;
    for (int tt = 0; tt < 8; ++tt) {
        const int jT  = cbase + tt;
        const int col = jT * 16 + l16;
        v8f cr = {};
        v8f ci = {};
        unsigned o = (unsigned)(half * 512 + col);
        for (int k0 = 0; k0 < 256; k0 += 4, o += 1024) {
            v2f ar = *(const v2f*)&sTr[arow + k0];
            v2f ai = *(const v2f*)&sTi[arow + k0];
            v2f nai = -ai;
            v2f br, bi;
            br.x = UHr[o];
            br.y = UHr[o + 256];
            bi.x = UHi[o];
            bi.y = UHi[o + 256];
            cr = WMMA_F32(ar,  br, cr);
            cr = WMMA_F32(nai, bi, cr);
            ci = WMMA_F32(ar,  bi, ci);
            ci = WMMA_F32(ai,  br, ci);
        }
        // complex64 (re,im) interleaved; non-temporal: output is write-once
        for (int v = 0; v < 8; ++v) {
            const int m = rT * 16 + v + half * 8;
            const int j = jT * 16 + l16;
            v2f val;
            val.x = cr[v];
            val.y = ci[v];
            __builtin_nontemporal_store(val, obase + m * 256 + j);
        }
    }
}

extern "C" void kernel_launch(void* const* d_in, const int* in_sizes, int n_in,
                              void* d_out, int out_size, void* d_ws, size_t ws_size,
                              hipStream_t stream)
{
    const float* rho_r = (const float*)d_in[0];   // (256, 256, 256) f32
    const float* rho_i = (const float*)d_in[1];   // (256, 256, 256) f32
    const float* phi_x = (const float*)d_in[2];   // (56,) f32
    const float* phi_y = (const float*)d_in[3];   // (56,) f32
    float* out = (float*)d_out;                   // (256,256,256) complex64 -> 2x f32

    float* Ur  = (float*)d_ws;                    // 65536 floats each
    float* Ui  = Ur  + 65536;
    float* UHr = Ui  + 65536;
    float* UHi = UHr + 65536;

    qconv_build_u<<<dim3(1, 1, 1), dim3(256, 1, 1), 0, stream>>>(
        phi_x, phi_y, Ur, Ui, UHr, UHi);

    const size_t shmem = (size_t)4 * RB * PITCH * sizeof(float);   // 266240 B
    qconv_apply<<<dim3(4, 256, 1), dim3(256, 1, 1), shmem, stream>>>(
        rho_r, rho_i, Ur, Ui, UHr, UHi, out);
}